// CRF_45621142618408
// MI455X (gfx1250) — compile-verified
//
#include <hip/hip_runtime.h>
#include <hip/hip_bf16.h>

// Viterbi CRF decode on gfx1250 (MI455X), wave32.
// One wave (32 lanes) per batch chain; lane j owns tag j.
// Forward step uses V_WMMA_F32_16X16X4_F32 as an exact outer-sum:
//   D[m][n] = state[m]*1 + T[m][n]   (single fma == single RNE add, matches f32 ref)
// The loop carries the masked WMMA A-operands directly (A0x/A1x), so the serial
// chain is: A -> wmma -> tree-argmax (depth 4) -> bpermute combine -> add/select -> A.
// Tree reduce is left-wins-ties over index-ordered leaves == jnp.argmax first-occurrence.
// All selects are branch-free so EXEC stays all-ones around every WMMA.
// Backpointers live in LDS; backtrack + one-hot write happens in-wave.

typedef __attribute__((ext_vector_type(2))) float v2f;
typedef __attribute__((ext_vector_type(8))) float v8f;

#define CRF_T 2048
#define CRF_K 32
#define CRF_CH 8  // timestep prefetch chunk

__device__ __forceinline__ v8f wmma_outer_sum(v2f a, v2f b, v8f c) {
    // D = A(16x4) x B(4x16) + C(16x16), fp32
    return __builtin_amdgcn_wmma_f32_16x16x4_f32(
        /*neg_a=*/false, a, /*neg_b=*/false, b,
        /*c_mod=*/(short)0, c, /*reuse_a=*/false, /*reuse_b=*/false);
}

// Tree max/argmax over two D tiles whose per-lane rows are index-ordered:
// dA rows {ibaseA + 0..7}, dB rows {ibaseB + 0..7}, all ibaseA+r < ibaseB+r'.
// Left-wins-ties (strict '>') == first-occurrence argmax.
__device__ __forceinline__ void tree_argmax16(const v8f& dA, const v8f& dB,
                                              int ibaseA, int ibaseB,
                                              float& outV, int& outI) {
    float tv[16]; int ti[16];
#pragma unroll
    for (int r = 0; r < 8; ++r) { tv[r]     = dA[r]; ti[r]     = ibaseA + r; }
#pragma unroll
    for (int r = 0; r < 8; ++r) { tv[8 + r] = dB[r]; ti[8 + r] = ibaseB + r; }
#pragma unroll
    for (int s = 1; s < 16; s <<= 1) {
#pragma unroll
        for (int k = 0; k < 16; k += 2 * s) {
            bool tk = tv[k + s] > tv[k];            // strict: left (smaller idx) wins ties
            tv[k] = tk ? tv[k + s] : tv[k];
            ti[k] = tk ? ti[k + s] : ti[k];
        }
    }
    outV = tv[0]; outI = ti[0];
}

__global__ __launch_bounds__(32) void crf_viterbi_kernel(
    const float* __restrict__ inputs,       // [B, T, K]
    const float* __restrict__ transitions,  // [K, K]
    float* __restrict__ out)                // [B, T, K] one-hot
{
    __shared__ unsigned char bpLds[(CRF_T - 1) * CRF_K];  // 65504 bytes

    const int b    = blockIdx.x;
    const int lane = threadIdx.x;            // 0..31
    const bool lo  = (lane < 16);

    const float* __restrict__ pin  = inputs + (size_t)b * CRF_T * CRF_K;
    float* __restrict__       pout = out    + (size_t)b * CRF_T * CRF_K;

    // ---- Load transition tiles in WMMA C/D layout ----
    // D layout (16x16 f32): VGPR r -> lanes 0-15: (M=r, N=lane); lanes 16-31: (M=r+8, N=lane-16)
    const int rb = lo ? 0 : 8;          // row base within tile for this lane
    const int cc = lane & 15;           // column within tile
    v8f c00, c10, c01, c11;
#pragma unroll
    for (int r = 0; r < 8; ++r) {
        c00[r] = transitions[(     rb + r) * CRF_K + (     cc)];  // rows 0-15,  cols 0-15
        c10[r] = transitions[(16 + rb + r) * CRF_K + (     cc)];  // rows 16-31, cols 0-15
        c01[r] = transitions[(     rb + r) * CRF_K + (16 + cc)];  // rows 0-15,  cols 16-31
        c11[r] = transitions[(16 + rb + r) * CRF_K + (16 + cc)];  // rows 16-31, cols 16-31
    }

    // B matrix (4x16): only row K=0 nonzero (all ones). K=0 occupies VGPR0 lanes 0-15.
    const v2f Bm = { lo ? 1.0f : 0.0f, 0.0f };

    // ---- init from potentials[:,0]; carry masked A-operand slices directly ----
    // A0x (rows 0-15 slice): lane m<16 holds state[m];   lanes>=16 hold 0 (K=2 col)
    // A1x (rows 16-31 slice): lane m<16 holds state[16+m]; lanes>=16 hold 0
    float s0init = pin[lane];
    float sSwap  = __shfl_xor(s0init, 16, 32);
    float A0x = lo ? s0init : 0.0f;
    float A1x = lo ? sSwap  : 0.0f;

    // One Viterbi step at time tt consuming potential `pot`.
    auto step = [&](float pot, int tt) {
        // swapped potential: independent of the state chain, scheduled early
        float potSwap = __shfl_xor(pot, 16, 32);

        v2f A0 = { A0x, 0.0f };     // rows 0-15
        v2f A1 = { A1x, 0.0f };     // rows 16-31

        // ---- 4 exact outer-sum WMMAs: scores[i][j] = state[i] + T[i][j] ----
        v8f d00 = wmma_outer_sum(A0, Bm, c00);
        v8f d10 = wmma_outer_sum(A1, Bm, c10);
        v8f d01 = wmma_outer_sum(A0, Bm, c01);
        v8f d11 = wmma_outer_sum(A1, Bm, c11);

        // ---- per-lane tree max/argmax over its 16 rows, per column group ----
        // group 0: column (lane&15), rows {rb..rb+7} (d00) and {16+rb..16+rb+7} (d10)
        // group 1: column 16+(lane&15), same rows from d01/d11
        float v0, v1; int i0, i1;
        tree_argmax16(d00, d10, rb, 16 + rb, v0, i0);
        tree_argmax16(d01, d11, rb, 16 + rb, v1, i1);

        // ---- combine with partner lane (rows interleave 0-7/16-23 vs 8-15/24-31) ----
        // tie-break: smaller source index wins (jnp.argmax first-occurrence).
        {
            float pv = __shfl_xor(v0, 16, 32); int pi = __shfl_xor(i0, 16, 32);
            bool tk = (bool)((pv > v0) | ((pv == v0) & (pi < i0)));
            v0 = tk ? pv : v0; i0 = tk ? pi : i0;
        }
        {
            float pv = __shfl_xor(v1, 16, 32); int pi = __shfl_xor(i1, 16, 32);
            bool tk = (bool)((pv > v1) | ((pv == v1) & (pi < i1)));
            v1 = tk ? pv : v1; i1 = tk ? pi : i1;
        }

        // After the symmetric combine, every lane holds:
        //   (v0,i0) = result for column (lane&15)      [group 0]
        //   (v1,i1) = result for column 16+(lane&15)   [group 1]
        // next A0x: lane m<16 needs state[m]    = v0[m] + pot[m]
        // next A1x: lane m<16 needs state[16+m] = v1[m] + potSwap[m]
        A0x = lo ? (v0 + pot)     : 0.0f;
        A1x = lo ? (v1 + potSwap) : 0.0f;

        int bestI = lo ? i0 : i1;
        bpLds[(tt - 1) * CRF_K + lane] = (unsigned char)bestI;
    };

    // ---- software-pipelined potential prefetch ----
    float p[CRF_CH];
#pragma unroll
    for (int j = 0; j < CRF_CH; ++j)
        p[j] = pin[(size_t)(1 + j) * CRF_K + lane];  // 1+j <= 8 < T always

    int tb = 1;
    // Steady state: full chunk, unguarded prefetch (needs tb + 2*CH - 1 <= T-1)
    for (; tb <= CRF_T - 2 * CRF_CH; tb += CRF_CH) {
        float np[CRF_CH];
#pragma unroll
        for (int j = 0; j < CRF_CH; ++j)
            np[j] = pin[(size_t)(tb + CRF_CH + j) * CRF_K + lane];
#pragma unroll
        for (int j = 0; j < CRF_CH; ++j)
            step(p[j], tb + j);
#pragma unroll
        for (int j = 0; j < CRF_CH; ++j) p[j] = np[j];
    }

    // One full chunk with guarded prefetch (covers the partial tail prefetch)
    if (tb + CRF_CH <= CRF_T) {
        float np[CRF_CH];
#pragma unroll
        for (int j = 0; j < CRF_CH; ++j) {
            int tt = tb + CRF_CH + j;
            np[j] = (tt < CRF_T) ? pin[(size_t)tt * CRF_K + lane] : 0.0f;
        }
#pragma unroll
        for (int j = 0; j < CRF_CH; ++j)
            step(p[j], tb + j);
#pragma unroll
        for (int j = 0; j < CRF_CH; ++j) p[j] = np[j];
        tb += CRF_CH;
    }

    // Epilogue: remaining (T-1) % CH steps (uniform guards)
#pragma unroll
    for (int j = 0; j < CRF_CH; ++j) {
        int tt = tb + j;
        if (tt < CRF_T) step(p[j], tt);
    }

    // ---- reconstruct full final state across 32 lanes ----
    // lane j<16: state[j] = A0x[j]; lane j>=16: state[j] = A1x[j-16]
    float state = lo ? A0x : __shfl_xor(A1x, 16, 32);

    // ---- last_tag = argmax_j(final_state), first-occurrence tie-break ----
    float v = state; int idx = lane;
#pragma unroll
    for (int off = 16; off >= 1; off >>= 1) {
        float pv = __shfl_xor(v, off, 32);
        int   pi = __shfl_xor(idx, off, 32);
        bool tk = (bool)((pv > v) | ((pv == v) & (pi < idx)));
        v = tk ? pv : v; idx = tk ? pi : idx;
    }
    int cur = idx;  // identical in all lanes after full butterfly

    // ---- backtrack through LDS backpointers; write one-hot rows ----
    pout[(size_t)(CRF_T - 1) * CRF_K + lane] = (lane == cur) ? 1.0f : 0.0f;
    for (int t = CRF_T - 2; t >= 0; --t) {
        cur = (int)bpLds[t * CRF_K + cur];  // uniform LDS broadcast read
        pout[(size_t)t * CRF_K + lane] = (lane == cur) ? 1.0f : 0.0f;
    }
}

extern "C" void kernel_launch(void* const* d_in, const int* in_sizes, int n_in,
                              void* d_out, int out_size, void* d_ws, size_t ws_size,
                              hipStream_t stream) {
    const float* inputs      = (const float*)d_in[0];  // [B, T, K] f32
    const float* transitions = (const float*)d_in[1];  // [K, K] f32
    float* out               = (float*)d_out;          // [B, T, K] f32 one-hot

    const int numB = in_sizes[0] / (CRF_T * CRF_K);    // = 512
    crf_viterbi_kernel<<<dim3(numB), dim3(32), 0, stream>>>(inputs, transitions, out);
}